// HybridANFIS_29978871726295
// MI455X (gfx1250) — compile-verified
//
#include <hip/hip_runtime.h>
#include <hip/hip_bf16.h>

// Problem constants (from setup_inputs): B=1024, D=16, M=4, R=4096, C=64
#define ANFIS_B 1024
#define ANFIS_D 16
#define ANFIS_M 4
#define ANFIS_R 4096
#define ANFIS_C 64
#define ANFIS_EPS 1e-9f

#define GEMM_WAVES 8        // waves per block, each owns a 16-row b-tile
#define GEMM_KSPLIT 8       // K split across blockIdx.y, partials atomically added
#define GEMM_KCHUNK (ANFIS_R / GEMM_KSPLIT)  // 512
#define GEMM_KTILE 64       // K panel staged in LDS per iteration (64x64 f32 = 16 KB)

typedef __attribute__((ext_vector_type(2))) float v2f;
typedef __attribute__((ext_vector_type(8))) float v8f;

// ---------------------------------------------------------------------------
// Kernel 1: cons_sum[r,c] = sum_j consequents[r,j,c]   (j = 0..16)
// ---------------------------------------------------------------------------
__global__ __launch_bounds__(256) void anfis_cons_sum_kernel(
    const float* __restrict__ cons, float* __restrict__ cons_sum) {
  int idx = blockIdx.x * blockDim.x + threadIdx.x;  // idx = r*C + c
  if (idx >= ANFIS_R * ANFIS_C) return;
  int r = idx >> 6;
  int c = idx & (ANFIS_C - 1);
  const float* p = cons + (size_t)r * (ANFIS_D + 1) * ANFIS_C + c;
  float s = 0.0f;
#pragma unroll
  for (int j = 0; j < ANFIS_D + 1; ++j) s += p[(size_t)j * ANFIS_C];
  cons_sum[idx] = s;
}

// ---------------------------------------------------------------------------
// Kernel 2: pack rules[r, 0..15] (values 0..3) into one u32 per rule.
// ---------------------------------------------------------------------------
__global__ __launch_bounds__(256) void anfis_pack_rules_kernel(
    const int* __restrict__ rules, unsigned* __restrict__ prule) {
  int r = blockIdx.x * blockDim.x + threadIdx.x;
  if (r >= ANFIS_R) return;
  unsigned w = 0u;
#pragma unroll
  for (int d = 0; d < ANFIS_D; ++d)
    w |= ((unsigned)rules[r * ANFIS_D + d] & 3u) << (2 * d);
  prule[r] = w;
}

// ---------------------------------------------------------------------------
// Kernel 3: x_ext[b,:] = [x[b,:], 1] (output #3), s_x[b] = sum(x_ext[b,:]),
// and zero-init of rule_out (required: harness poisons d_out, and the GEMM
// accumulates with atomics).
// ---------------------------------------------------------------------------
__global__ __launch_bounds__(256) void anfis_xext_zero_kernel(
    const float* __restrict__ x, float* __restrict__ x_ext,
    float* __restrict__ s_x, float* __restrict__ rule_out) {
  int i = blockIdx.x * blockDim.x + threadIdx.x;
  if (i < ANFIS_B * ANFIS_C) rule_out[i] = 0.0f;
  if (i >= ANFIS_B) return;
  int b = i;
  float s = 1.0f;
#pragma unroll
  for (int d = 0; d < ANFIS_D; ++d) {
    float v = x[b * ANFIS_D + d];
    x_ext[b * (ANFIS_D + 1) + d] = v;
    s += v;
  }
  x_ext[b * (ANFIS_D + 1) + ANFIS_D] = 1.0f;
  s_x[b] = s;
}

// ---------------------------------------------------------------------------
// Kernel 4: one block per batch row b.
//   - threads 0..63 build mfs[d][m] in LDS (Gaussian MF + EPS)
//   - each of 256 threads computes 16 rule firing strengths via packed-rule
//     LDS gathers, block-reduces the row sum, normalizes, writes nfs[b,:].
// ---------------------------------------------------------------------------
__global__ __launch_bounds__(256) void anfis_fuzz_kernel(
    const float* __restrict__ x, const float* __restrict__ centers,
    const float* __restrict__ widths, const unsigned* __restrict__ prule,
    float* __restrict__ nfs) {
  __shared__ float mfs[ANFIS_D * ANFIS_M];  // 64 floats
  __shared__ float red[256];
  const int b = blockIdx.x;
  const int t = threadIdx.x;

  if (t < ANFIS_D * ANFIS_M) {
    int d = t >> 2, m = t & 3;
    float xv = x[b * ANFIS_D + d];
    float c  = centers[d * ANFIS_M + m];
    float w  = widths[d * ANFIS_M + m];
    float diff = xv - c;
    mfs[t] = __expf(-(diff * diff) / (2.0f * w * w)) + ANFIS_EPS;
  }
  __syncthreads();

  float fs[ANFIS_R / 256];
  float lsum = 0.0f;
#pragma unroll
  for (int i = 0; i < ANFIS_R / 256; ++i) {
    int r = t + i * 256;
    unsigned w = prule[r];
    float p = 1.0f;
#pragma unroll
    for (int d = 0; d < ANFIS_D; ++d)
      p *= mfs[(d << 2) | ((w >> (2 * d)) & 3u)];
    fs[i] = p;
    lsum += p;
  }
  red[t] = lsum;
  __syncthreads();
#pragma unroll
  for (int s = 128; s > 0; s >>= 1) {
    if (t < s) red[t] += red[t + s];
    __syncthreads();
  }
  float inv = 1.0f / (red[0] + ANFIS_EPS);
#pragma unroll
  for (int i = 0; i < ANFIS_R / 256; ++i)
    nfs[(size_t)b * ANFIS_R + t + i * 256] = fs[i] * inv;
}

// ---------------------------------------------------------------------------
// Kernel 5: rule_out[b,c] += s_x[b] * sum_{r in chunk} nfs[b,r]*cons_sum[r,c]
//
// fp32 GEMM via V_WMMA_F32_16X16X4_F32 (wave32), 8 waves per block sharing a
// cons_sum K-panel staged in LDS with GLOBAL_LOAD_ASYNC_TO_LDS_B128
// (ASYNCcnt path, inline asm per CDNA5 ISA). Each wave
// owns one 16-row b-tile x all 64 columns (4 f32 accumulators). K is split
// across blockIdx.y; partials (pre-scaled by s_x, which distributes over the
// linear sum) are accumulated with global_atomic_add_f32.
//
// ISA §7.12.2 layouts (32-bit):
//   A 16x4 : lanes 0-15 -> row M=lane, VGPR{0,1}=K{0,1}; lanes 16-31 K{2,3}
//   B 4x16 : lanes 0-15 -> col N=lane, VGPR{0,1}=K{0,1}; lanes 16-31 K{2,3}
//   C/D    : VGPR v -> row v (lanes 0-15) / row v+8 (lanes 16-31), N=lane%16
// ---------------------------------------------------------------------------
__global__ __launch_bounds__(32 * GEMM_WAVES) void anfis_gemm_wmma_kernel(
    const float* __restrict__ nfs,       // [B, R]
    const float* __restrict__ cons_sum,  // [R, C]
    const float* __restrict__ s_x,       // [B]
    float* __restrict__ rule_out) {      // [B, C] (accumulated)
  __shared__ float Bs[GEMM_KTILE * ANFIS_C];  // 16 KB K-panel of cons_sum

  const int tid  = threadIdx.x;
  const int wave = tid >> 5;
  const int lane = tid & 31;
  const int half = lane >> 4;  // selects K pair within the 16x16x4 fragment
  const int l16  = lane & 15;

  const int b0 = (blockIdx.x * GEMM_WAVES + wave) * 16;  // b-tile row base
  const int kc = blockIdx.y * GEMM_KCHUNK;               // K chunk base

  v8f acc[4] = {v8f{}, v8f{}, v8f{}, v8f{}};

  // A fragment source: row b0+l16, K-pair base 2*half (contiguous -> b64)
  const float* arow = nfs + (size_t)(b0 + l16) * ANFIS_R + 2 * half + kc;

  for (int kt = 0; kt < GEMM_KCHUNK; kt += GEMM_KTILE) {
    // --- Stage cons_sum[kc+kt .. +KTILE) x 64 into LDS asynchronously ----
    // 256 threads x 4 iters x 16B = 16 KB.
#pragma unroll
    for (int i = 0; i < (GEMM_KTILE * ANFIS_C) / (256 * 4); ++i) {
      int idx = tid + i * 256;  // float4 index within the panel
      const float* gsrc = cons_sum + (size_t)(kc + kt) * ANFIS_C + idx * 4;
      unsigned ldsdst = (unsigned)(uintptr_t)(Bs + idx * 4);  // LDS byte off
      asm volatile("global_load_async_to_lds_b128 %0, %1, off"
                   :: "v"(ldsdst), "v"(gsrc) : "memory");
    }
    asm volatile("s_wait_asynccnt 0x0" ::: "memory");
    __syncthreads();

    // Prefetch the next A cacheline while we compute on this panel.
    __builtin_prefetch(arow + kt + GEMM_KTILE, 0, 1);

    // --- 16 k-steps of 4 on the LDS-resident panel -----------------------
#pragma unroll 4
    for (int k2 = 0; k2 < GEMM_KTILE; k2 += 4) {
      v2f a;
      a.x = arow[kt + k2 + 0];
      a.y = arow[kt + k2 + 1];
      const float* brow0 = Bs + (k2 + 2 * half) * ANFIS_C;
      const float* brow1 = brow0 + ANFIS_C;
#pragma unroll
      for (int tcol = 0; tcol < 4; ++tcol) {
        v2f bb;
        bb.x = brow0[tcol * 16 + l16];
        bb.y = brow1[tcol * 16 + l16];
        // (neg_a, A, neg_b, B, c_mod, C, reuse_a, reuse_b)
        acc[tcol] = __builtin_amdgcn_wmma_f32_16x16x4_f32(
            false, a, false, bb, (short)0, acc[tcol], false, false);
      }
    }
    __syncthreads();  // panel reuse barrier before next stage
  }

  // Epilogue: scale partial by s_x (linearity) and atomically accumulate.
#pragma unroll
  for (int v = 0; v < 8; ++v) {
    int row = b0 + v + 8 * half;
    float s = s_x[row];
#pragma unroll
    for (int tcol = 0; tcol < 4; ++tcol) {
      atomicAdd(&rule_out[(size_t)row * ANFIS_C + tcol * 16 + l16],
                acc[tcol][v] * s);
    }
  }
}

// ---------------------------------------------------------------------------
// Launch: inputs {x, centers, widths, consequents, rules}; outputs
// rule_out | nfs | x_ext concatenated flat in d_out.
// ---------------------------------------------------------------------------
extern "C" void kernel_launch(void* const* d_in, const int* in_sizes, int n_in,
                              void* d_out, int out_size, void* d_ws,
                              size_t ws_size, hipStream_t stream) {
  const float* x       = (const float*)d_in[0];
  const float* centers = (const float*)d_in[1];
  const float* widths  = (const float*)d_in[2];
  const float* cons    = (const float*)d_in[3];
  const int*   rules   = (const int*)d_in[4];

  float* rule_out = (float*)d_out;                         // B*C
  float* nfs      = rule_out + (size_t)ANFIS_B * ANFIS_C;  // B*R
  float* x_ext    = nfs + (size_t)ANFIS_B * ANFIS_R;       // B*(D+1)

  // Workspace: cons_sum (1 MB) | packed rules (16 KB) | s_x (4 KB)
  float*    cons_sum = (float*)d_ws;
  unsigned* prule    = (unsigned*)(cons_sum + (size_t)ANFIS_R * ANFIS_C);
  float*    s_x      = (float*)(prule + ANFIS_R);

  anfis_cons_sum_kernel<<<(ANFIS_R * ANFIS_C + 255) / 256, 256, 0, stream>>>(
      cons, cons_sum);
  anfis_pack_rules_kernel<<<(ANFIS_R + 255) / 256, 256, 0, stream>>>(
      rules, prule);
  anfis_xext_zero_kernel<<<(ANFIS_B * ANFIS_C + 255) / 256, 256, 0, stream>>>(
      x, x_ext, s_x, rule_out);
  anfis_fuzz_kernel<<<ANFIS_B, 256, 0, stream>>>(
      x, centers, widths, prule, nfs);

  dim3 ggrid(ANFIS_B / (16 * GEMM_WAVES), GEMM_KSPLIT);  // (8, 8)
  anfis_gemm_wmma_kernel<<<ggrid, 32 * GEMM_WAVES, 0, stream>>>(
      nfs, cons_sum, s_x, rule_out);
}